// VanillaGNN_1855425872280
// MI455X (gfx1250) — compile-verified
//
#include <hip/hip_runtime.h>
#include <hip/hip_bf16.h>

// ---------------------------------------------------------------------------
// GCN forward (2x GCNConv + linear classifier) for MI455X / gfx1250, wave32.
// Dense matmuls use V_WMMA_F32_16X16X4_F32 (f32 in/out, 16x16 tile per wave).
// Edge aggregation uses coalesced float4 gathers + global_atomic_add_f32.
// ---------------------------------------------------------------------------

typedef float v2f __attribute__((ext_vector_type(2)));
typedef float v8f __attribute__((ext_vector_type(8)));

#define D_FEAT 128

// --- degree / normalization --------------------------------------------------

__global__ void k_init_deg(float* __restrict__ deg, int n) {
  int t = blockIdx.x * blockDim.x + threadIdx.x;
  if (t < n) deg[t] = 1.0f;  // self loop
}

__global__ void k_accum_deg(const int* __restrict__ dst, float* __restrict__ deg, int e) {
  int t = blockIdx.x * blockDim.x + threadIdx.x;
  if (t < e) atomicAdd(&deg[dst[t]], 1.0f);
}

__global__ void k_inv_sqrt(float* __restrict__ deg, int n) {
  int t = blockIdx.x * blockDim.x + threadIdx.x;
  if (t < n) {
    float d = deg[t];
    deg[t] = (d > 0.0f) ? __frsqrt_rn(d) : 0.0f;
  }
}

// --- dense GEMM: H[n,128] = X[n,128] @ W[128,128] ---------------------------
// One wave32 per 16x16 output tile. blockDim = 256 (8 waves); wave w handles
// column tile w (cols 16w..16w+15). gridDim.x = n/16 row tiles (n % 16 == 0).
// K loop: 128 in steps of 4 via V_WMMA_F32_16X16X4_F32.
//
// A fragment (16x4, f32): lanes 0-15 -> M=lane, {K, K+1}; lanes 16-31 -> {K+2, K+3}.
// B fragment (4x16, f32): lanes 0-15 -> N=lane, {K, K+1}; lanes 16-31 -> {K+2, K+3}.
// C/D (16x16 f32, 8 VGPRs): VGPR j -> row j (lanes 0-15) / row j+8 (lanes 16-31).

__global__ __launch_bounds__(256) void k_gemm128(const float* __restrict__ X,
                                                 const float* __restrict__ W,
                                                 float* __restrict__ H) {
  const int lane  = threadIdx.x & 31;
  const int wave  = threadIdx.x >> 5;          // 0..7 : column tile
  const int row0  = blockIdx.x << 4;           // 16 rows per tile
  const int col0  = wave << 4;                 // 16 cols per tile
  const int m     = lane & 15;                 // row (A) / col (B) within tile
  const int khalf = (lane >> 4) << 1;          // 0 for lanes 0-15, 2 for 16-31

  const float* __restrict__ xrow = X + (size_t)(row0 + m) * D_FEAT;
  v8f acc = {};

#pragma unroll 4
  for (int k = 0; k < D_FEAT; k += 4) {
    v2f a, b;
    a.x = xrow[k + khalf];
    a.y = xrow[k + khalf + 1];
    b.x = W[(size_t)(k + khalf) * D_FEAT + col0 + m];
    b.y = W[(size_t)(k + khalf + 1) * D_FEAT + col0 + m];
    acc = __builtin_amdgcn_wmma_f32_16x16x4_f32(
        /*neg_a=*/false, a, /*neg_b=*/false, b,
        /*c_mod=*/(short)0, acc, /*reuse_a=*/false, /*reuse_b=*/false);
  }

  // VGPR j holds row (j + 8*(lane>>4)), column (col0 + m)
  float* __restrict__ hp =
      H + (size_t)(row0 + ((lane >> 4) << 3)) * D_FEAT + col0 + m;
#pragma unroll
  for (int j = 0; j < 8; ++j) hp[(size_t)j * D_FEAT] = acc[j];
}

// --- zero accumulation buffer ----------------------------------------------

__global__ void k_zero(float* __restrict__ p, size_t n) {
  size_t t = (size_t)blockIdx.x * blockDim.x + threadIdx.x;
  if (t < n) p[t] = 0.0f;
}

// --- edge scatter: out[dst] += h[src] * inv[src]*inv[dst] -------------------
// One wave32 per edge; each lane moves a float4 (coalesced 512B gather) and
// issues 4 global_atomic_add_f32 (no return -> STOREcnt only).

__global__ __launch_bounds__(256) void k_scatter(const int* __restrict__ src,
                                                 const int* __restrict__ dst,
                                                 const float* __restrict__ inv,
                                                 const float* __restrict__ h,
                                                 float* __restrict__ out, int e) {
  int edge = blockIdx.x * 8 + (threadIdx.x >> 5);
  if (edge >= e) return;
  const int lane = threadIdx.x & 31;
  const int s = src[edge];
  const int d = dst[edge];
  const float nrm = inv[s] * inv[d];

  const float4 v = ((const float4*)(h + (size_t)s * D_FEAT))[lane];
  float* __restrict__ op = out + (size_t)d * D_FEAT + lane * 4;
  atomicAdd(op + 0, v.x * nrm);
  atomicAdd(op + 1, v.y * nrm);
  atomicAdd(op + 2, v.z * nrm);
  atomicAdd(op + 3, v.w * nrm);
}

// --- self loop + bias + relu epilogue ---------------------------------------
// out[i,d] = relu(out[i,d] + hraw[i,d]*inv[i]^2 + b[d])

__global__ void k_finish(const float* __restrict__ hraw,
                         const float* __restrict__ inv,
                         const float* __restrict__ b,
                         float* __restrict__ out, size_t total) {
  size_t t = (size_t)blockIdx.x * blockDim.x + threadIdx.x;
  if (t >= total) return;
  int i = (int)(t >> 7);
  int d = (int)(t & (D_FEAT - 1));
  float w = inv[i];
  float v = out[t] + hraw[t] * w * w + b[d];
  out[t] = v > 0.0f ? v : 0.0f;
}

// --- classifier: logits[n,2] = H[n,128] @ Wc[128,2] + bc --------------------

__global__ void k_classifier(const float* __restrict__ h,
                             const float* __restrict__ Wc,
                             const float* __restrict__ bc,
                             float* __restrict__ out, int n) {
  int i = blockIdx.x * blockDim.x + threadIdx.x;
  if (i >= n) return;
  const float* __restrict__ hp = h + (size_t)i * D_FEAT;
  float a0 = bc[0], a1 = bc[1];
#pragma unroll 8
  for (int k = 0; k < D_FEAT; ++k) {
    float hv = hp[k];
    a0 += hv * Wc[k * 2 + 0];
    a1 += hv * Wc[k * 2 + 1];
  }
  out[(size_t)i * 2 + 0] = a0;
  out[(size_t)i * 2 + 1] = a1;
}

// ---------------------------------------------------------------------------

extern "C" void kernel_launch(void* const* d_in, const int* in_sizes, int n_in,
                              void* d_out, int out_size, void* d_ws, size_t ws_size,
                              hipStream_t stream) {
  const float* x   = (const float*)d_in[0];
  const int*   ei  = (const int*)d_in[1];     // [2, E] int32
  const float* W1  = (const float*)d_in[2];
  const float* b1  = (const float*)d_in[3];
  const float* W2  = (const float*)d_in[4];
  const float* b2  = (const float*)d_in[5];
  const float* Wc  = (const float*)d_in[6];
  const float* bc  = (const float*)d_in[7];
  float* out = (float*)d_out;

  const int n = in_sizes[0] / D_FEAT;         // 50000 (multiple of 16)
  const int e = in_sizes[1] / 2;              // 800000
  const int* src = ei;
  const int* dst = ei + e;

  const size_t feat_elems = (size_t)n * D_FEAT;

  // Workspace layout: inv[n] | bufA[n*128] | bufB[n*128]
  float* inv  = (float*)d_ws;
  float* bufA = inv + n;
  float* bufB = bufA + feat_elems;

  const int T = 256;
  const int gN    = (n + T - 1) / T;
  const int gE    = (e + T - 1) / T;
  const int gFeat = (int)((feat_elems + T - 1) / T);
  const int gTile = n / 16;                   // 3125 row tiles
  const int gScat = (e + 7) / 8;              // 8 edges (waves) per block

  // degree -> inv_sqrt
  k_init_deg<<<gN, T, 0, stream>>>(inv, n);
  k_accum_deg<<<gE, T, 0, stream>>>(dst, inv, e);
  k_inv_sqrt<<<gN, T, 0, stream>>>(inv, n);

  // ---- layer 1 ----
  k_gemm128<<<gTile, T, 0, stream>>>(x, W1, bufA);
  k_zero<<<gFeat, T, 0, stream>>>(bufB, feat_elems);
  k_scatter<<<gScat, T, 0, stream>>>(src, dst, inv, bufA, bufB, e);
  k_finish<<<gFeat, T, 0, stream>>>(bufA, inv, b1, bufB, feat_elems);

  // ---- layer 2 ----
  k_gemm128<<<gTile, T, 0, stream>>>(bufB, W2, bufA);
  k_zero<<<gFeat, T, 0, stream>>>(bufB, feat_elems);
  k_scatter<<<gScat, T, 0, stream>>>(src, dst, inv, bufA, bufB, e);
  k_finish<<<gFeat, T, 0, stream>>>(bufA, inv, b2, bufB, feat_elems);

  // ---- classifier ----
  k_classifier<<<gN, T, 0, stream>>>(bufB, Wc, bc, out, n);
}